// Loss_84164179132563
// MI455X (gfx1250) — compile-verified
//
#include <hip/hip_runtime.h>

// Neighbor2Neighbor fused mask-pair + dual subimage gather for MI455X (gfx1250).
//
// Shapes (fixed by reference setup): img (32,3,1024,1024) f32,
// rd_idx (32,512,512) int (harness contract: integer inputs -> const int*),
// out (2,32,3,512,512) f32.
//
// Pure data movement: ~637 MB total traffic, 0 FLOPs -> HBM-bound,
// ~27 us floor at 23.3 TB/s. Strategy: 128-bit non-temporal loads/stores,
// one thread = 4 output pixels x 3 channels, index loaded once per thread.
// Grid encodes (i, nn) so thread-index linearization costs zero VALU.

typedef float f4 __attribute__((ext_vector_type(4)));
typedef int   i4 __attribute__((ext_vector_type(4)));

// IDX_PAIR collapsed to nibble LUTs: k1[idx] and k2[idx] for idx in 0..7.
// rows: (0,1)(0,2)(1,3)(2,3)(1,0)(2,0)(3,1)(3,2)
#define K1_TAB 0x33212100u
#define K2_TAB 0x21003321u

__device__ __forceinline__ float sel4(float p0, float p1, float p2, float p3, int k) {
    // k bit0: second element of the row pair; k bit1: bottom row pair.
    float a = (k & 1) ? p1 : p0;
    float b = (k & 1) ? p3 : p2;
    return (k & 2) ? b : a;
}

__global__ __launch_bounds__(128) void n2n_gather_kernel(
    const float* __restrict__ img,
    const int*   __restrict__ rd,
    float*       __restrict__ out)
{
    constexpr int  N = 32, C = 3, H = 1024, W = 1024;
    constexpr int  H2 = H / 2, W2 = W / 2;
    constexpr long OUT_HALF = (long)N * C * H2 * W2; // offset of subimage2

    // blockDim.x == W2/4 == 128: one block = one output row, all channels.
    const unsigned j4 = threadIdx.x;     // quad index along output width
    const unsigned i  = blockIdx.x;      // output row     (SGPR)
    const unsigned nn = blockIdx.y;      // batch          (SGPR)
    const unsigned j  = j4 << 2;         // output column base (4 outputs)

    // ---- index load: 4 x int32, 16B/lane, 512B contiguous per wave ----
    const i4 kidx = __builtin_nontemporal_load(
        (const i4*)(rd + ((long)nn * H2 + i) * W2 + j));

    int k1[4], k2[4];
#pragma unroll
    for (int r = 0; r < 4; ++r) {
        const int e = ((const int*)&kidx)[r] & 7;
        k1[r] = (int)((K1_TAB >> (4 * e)) & 3u);
        k2[r] = (int)((K2_TAB >> (4 * e)) & 3u);
    }

#pragma unroll
    for (int c = 0; c < C; ++c) {
        // Input 2x2 blocks: rows 2i and 2i+1, columns [2j, 2j+8).
        const float* row0 = img + (((long)(nn * C + c) * H + 2 * i) * W) + 2 * j;
        const float* row1 = row0 + W;

        const f4 a0 = __builtin_nontemporal_load((const f4*)(row0));
        const f4 a1 = __builtin_nontemporal_load((const f4*)(row0 + 4));
        const f4 b0 = __builtin_nontemporal_load((const f4*)(row1));
        const f4 b1 = __builtin_nontemporal_load((const f4*)(row1 + 4));

        // Block r pixels (row-major p0,p1,p2,p3):
        //  r=0: a0.x a0.y b0.x b0.y   r=1: a0.z a0.w b0.z b0.w
        //  r=2: a1.x a1.y b1.x b1.y   r=3: a1.z a1.w b1.z b1.w
        f4 o1, o2;
        o1.x = sel4(a0.x, a0.y, b0.x, b0.y, k1[0]);
        o1.y = sel4(a0.z, a0.w, b0.z, b0.w, k1[1]);
        o1.z = sel4(a1.x, a1.y, b1.x, b1.y, k1[2]);
        o1.w = sel4(a1.z, a1.w, b1.z, b1.w, k1[3]);

        o2.x = sel4(a0.x, a0.y, b0.x, b0.y, k2[0]);
        o2.y = sel4(a0.z, a0.w, b0.z, b0.w, k2[1]);
        o2.z = sel4(a1.x, a1.y, b1.x, b1.y, k2[2]);
        o2.w = sel4(a1.z, a1.w, b1.z, b1.w, k2[3]);

        const long o = ((long)(nn * C + c) * H2 + i) * W2 + j;
        __builtin_nontemporal_store(o1, (f4*)(out + o));
        __builtin_nontemporal_store(o2, (f4*)(out + OUT_HALF + o));
    }
}

extern "C" void kernel_launch(void* const* d_in, const int* in_sizes, int n_in,
                              void* d_out, int out_size, void* d_ws, size_t ws_size,
                              hipStream_t stream) {
    const float* img = (const float*)d_in[0];
    const int*   rd  = (const int*)d_in[1];
    float*       out = (float*)d_out;

    constexpr int N = 32, H2 = 512;
    // block = one output row of quads (128 lanes = 4 wave32s);
    // grid = (rows, batches) -> 16384 workgroups, no tail, no bounds check.
    dim3 block(128, 1, 1);
    dim3 grid(H2, N, 1);

    n2n_gather_kernel<<<grid, block, 0, stream>>>(img, rd, out);
}